// CapsuleLayer_1_69002944578131
// MI455X (gfx1250) — compile-verified
//
#include <hip/hip_runtime.h>

typedef float v2f __attribute__((ext_vector_type(2)));
typedef float v8f __attribute__((ext_vector_type(8)));

#define B_SZ      32
#define IN_CAPS   2048
#define IN_DIM    16
#define NUM_CAPS  32
#define DIM_CAPS  16
#define ROUT_EPS  1e-7f

#define WGS       256          // 8 wave32 waves per workgroup
#define G_BLOCKS  128
#define IPG       (IN_CAPS / G_BLOCKS)   // 16 input capsules per workgroup

// ---------------------------------------------------------------------------
// DPP sum across each 16-lane half (no LDS, no dscnt waits):
//   quad_perm[1,0,3,2] -> quad_perm[2,3,0,1] -> row_half_mirror -> row_mirror
// After this every lane in a 16-lane half holds the half's sum.
// ---------------------------------------------------------------------------
template <int CTRL>
__device__ __forceinline__ float dpp_add_step(float v) {
  return v + __int_as_float(__builtin_amdgcn_update_dpp(
      0, __float_as_int(v), CTRL, 0xF, 0xF, true));
}
__device__ __forceinline__ float half_row_sum16(float v) {
  v = dpp_add_step<0xB1>(v);    // quad_perm [1,0,3,2]  (xor 1)
  v = dpp_add_step<0x4E>(v);    // quad_perm [2,3,0,1]  (xor 2)
  v = dpp_add_step<0x141>(v);   // row_half_mirror      (quad swap in 8)
  v = dpp_add_step<0x140>(v);   // row_mirror           (8-group swap in 16)
  return v;
}

// ---------------------------------------------------------------------------
// Kernel 0: zero the [B, NUM_CAPS, DIM_CAPS] fp32 accumulator in workspace
// ---------------------------------------------------------------------------
__global__ __launch_bounds__(256) void caps_zero(float* __restrict__ acc) {
  int t = blockIdx.x * blockDim.x + threadIdx.x;
  if (t < B_SZ * NUM_CAPS * DIM_CAPS) acc[t] = 0.f;
}

// ---------------------------------------------------------------------------
// Kernel 1: fused  u = W·x  (WMMA f32 16x16x4)  +  3-round routing  +
//           accumulation of c2*u over input capsules (atomic into acc).
//
// Wave w owns tiles (btile in {0,1}) x (n in {w, w+8, w+16, w+24}).
// C/D tile layout (ISA 7.12.2): VGPR r, lanes 0-15 -> (M=r, N=lane),
// lanes 16-31 -> (M=r+8, N=lane-16).  M = batch row in tile, N = dim.
// A (16x4 f32): lanes 0-15 hold K=0,1 in VGPR0/1; lanes 16-31 hold K=2,3.
// B (4x16 f32): mirrored — lane%16 = column, K=0,1 / K=2,3 across halves.
// ---------------------------------------------------------------------------
__global__ __launch_bounds__(WGS) void caps_route(const float* __restrict__ x,
                                                  const float* __restrict__ W,
                                                  float* __restrict__ acc) {
  __shared__ float s_lds[B_SZ * NUM_CAPS];   // s[b][n] = sum_d u[b,i,n,d]
  __shared__ float c2_lds[B_SZ * NUM_CAPS];  // c2[b][n]

  const int tid  = threadIdx.x;
  const int lane = tid & 31;
  const int wave = tid >> 5;
  const int half = lane >> 4;    // 0: lanes 0-15, 1: lanes 16-31
  const int lq   = lane & 15;    // N index (= d) / A row within half

  v8f zero = {};
  v8f accv[8];
  #pragma unroll
  for (int t = 0; t < 8; ++t) accv[t] = zero;

  const int i0 = blockIdx.x * IPG;

  for (int ii = 0; ii < IPG; ++ii) {
    const int i = i0 + ii;

    // ---- A tiles: x[btile*16 + row, i, k..k+1] per the 16x4 f32 layout ----
    v2f a[2][4];
    #pragma unroll
    for (int bt = 0; bt < 2; ++bt) {
      const int brow = bt * 16 + lq;
      const float* xp = x + ((size_t)brow * IN_CAPS + i) * IN_DIM;
      #pragma unroll
      for (int ks = 0; ks < 4; ++ks) {
        a[bt][ks] = *(const v2f*)(xp + ks * 4 + half * 2);
      }
    }

    // ---- u tiles: for each owned n, B tile from W[i, n, d, k] ----
    v8f u[8];
    #pragma unroll
    for (int j = 0; j < 4; ++j) {
      const int n = wave + 8 * j;
      const float* wp = W + (((size_t)i * NUM_CAPS + n) * DIM_CAPS + lq) * IN_DIM;
      v2f bm[4];
      #pragma unroll
      for (int ks = 0; ks < 4; ++ks) {
        bm[ks] = *(const v2f*)(wp + ks * 4 + half * 2);
      }
      #pragma unroll
      for (int bt = 0; bt < 2; ++bt) {
        v8f c = {};
        #pragma unroll
        for (int ks = 0; ks < 4; ++ks) {
          c = __builtin_amdgcn_wmma_f32_16x16x4_f32(
                false, a[bt][ks], false, bm[ks], (short)0, c, false, false);
        }
        u[j * 2 + bt] = c;
      }
    }

    // ---- s[b][n] = sum over d: DPP reduce, then one store per tile ----
    #pragma unroll
    for (int t = 0; t < 8; ++t) {
      const int j = t >> 1, bt = t & 1;
      const int n = wave + 8 * j;
      float red[8];
      #pragma unroll
      for (int r = 0; r < 8; ++r) red[r] = half_row_sum16(u[t][r]);
      // binary select tree: lane lq (< 8) picks red[lq]
      const float r01 = (lq & 1) ? red[1] : red[0];
      const float r23 = (lq & 1) ? red[3] : red[2];
      const float r45 = (lq & 1) ? red[5] : red[4];
      const float r67 = (lq & 1) ? red[7] : red[6];
      const float r03 = (lq & 2) ? r23 : r01;
      const float r47 = (lq & 2) ? r67 : r45;
      const float rv  = (lq & 4) ? r47 : r03;
      if (lq < 8) {
        const int b = bt * 16 + lq + half * 8;
        s_lds[b * NUM_CAPS + n] = rv;
      }
    }
    __syncthreads();

    // ---- wave 0: per-batch routing (two softmaxes over 32 capsules) ----
    if (tid < 32) {
      const int b = tid;
      float s[32];
      #pragma unroll
      for (int n = 0; n < 32; ++n) s[n] = s_lds[b * NUM_CAPS + n];
      // c1 = softmax(s/32)
      float m1 = -3.4e38f;
      #pragma unroll
      for (int n = 0; n < 32; ++n) m1 = fmaxf(m1, s[n] * 0.03125f);
      float sum1 = 0.f;
      #pragma unroll
      for (int n = 0; n < 32; ++n) sum1 += __expf(s[n] * 0.03125f - m1);
      const float inv1 = 1.f / sum1;
      // b2 = s/32 + c1*s   (overwrite s)
      #pragma unroll
      for (int n = 0; n < 32; ++n) {
        const float b1 = s[n] * 0.03125f;
        s[n] = b1 + __expf(b1 - m1) * inv1 * s[n];
      }
      // c2 = softmax(b2)
      float m2 = -3.4e38f;
      #pragma unroll
      for (int n = 0; n < 32; ++n) m2 = fmaxf(m2, s[n]);
      float sum2 = 0.f;
      #pragma unroll
      for (int n = 0; n < 32; ++n) sum2 += __expf(s[n] - m2);
      const float inv2 = 1.f / sum2;
      #pragma unroll
      for (int n = 0; n < 32; ++n)
        c2_lds[b * NUM_CAPS + n] = __expf(s[n] - m2) * inv2;
    }
    __syncthreads();

    // ---- accumulate c2 * u into per-wave registers ----
    #pragma unroll
    for (int t = 0; t < 8; ++t) {
      const int j = t >> 1, bt = t & 1;
      const int n = wave + 8 * j;
      #pragma unroll
      for (int r = 0; r < 8; ++r) {
        const int b = bt * 16 + r + half * 8;
        accv[t][r] += c2_lds[b * NUM_CAPS + n] * u[t][r];
      }
    }
  }

  // ---- reduce across workgroups (distinct (b,n,d) owners within a WG) ----
  #pragma unroll
  for (int t = 0; t < 8; ++t) {
    const int j = t >> 1, bt = t & 1;
    const int n = wave + 8 * j;
    #pragma unroll
    for (int r = 0; r < 8; ++r) {
      const int b = bt * 16 + r + half * 8;
      atomicAdd(&acc[(b * NUM_CAPS + n) * DIM_CAPS + lq], accv[t][r]);
    }
  }
}

// ---------------------------------------------------------------------------
// Kernel 2: squash over the dim axis (groups of 16 lanes share one (b,n))
// ---------------------------------------------------------------------------
__global__ __launch_bounds__(256) void caps_squash(const float* __restrict__ acc,
                                                   float* __restrict__ out) {
  const int t = blockIdx.x * blockDim.x + threadIdx.x;   // 0..16383
  const float v = acc[t];
  const float s = half_row_sum16(v * v);
  const float scale = (s / (1.f + s)) * rsqrtf(s + ROUT_EPS);
  out[t] = scale * v;
}

// ---------------------------------------------------------------------------
extern "C" void kernel_launch(void* const* d_in, const int* in_sizes, int n_in,
                              void* d_out, int out_size, void* d_ws, size_t ws_size,
                              hipStream_t stream) {
  const float* x = (const float*)d_in[0];   // [32, 2048, 16]
  const float* W = (const float*)d_in[1];   // [1, 2048, 32, 16, 16]
  float* out = (float*)d_out;               // [32, 32, 16]
  float* acc = (float*)d_ws;                // 16384 floats (64 KB)

  const int NOUT = B_SZ * NUM_CAPS * DIM_CAPS;   // 16384
  caps_zero<<<(NOUT + 255) / 256, 256, 0, stream>>>(acc);
  caps_route<<<G_BLOCKS, WGS, 0, stream>>>(x, W, acc);
  caps_squash<<<NOUT / 256, 256, 0, stream>>>(acc, out);
}